// Appro_WAConv2d_32401233281026
// MI455X (gfx1250) — compile-verified
//
#include <hip/hip_runtime.h>
#include <hip/hip_bf16.h>
#include <stdint.h>

#define EPSF    1e-7f
#define NC      192
#define HH      56
#define WW      56
#define KK      49
#define PITCH   64            // padded LDS row pitch in floats
#define PROWS   62            // 56 + 2*3 halo rows
#define INTER   (HH * WW)     // 3136 interior pixels
#define NTASK   (WW * (HH/4)) // 784 four-row output strips (VALU fallback)
#define TPB     256
#define NTILE   14            // 224 (row, col-block) tasks / 16 per WMMA tile

#if __has_builtin(__builtin_amdgcn_tensor_store_from_lds)
#define HAVE_TDM 1
#else
#define HAVE_TDM 0
#endif

#if __has_builtin(__builtin_amdgcn_wmma_f32_16x16x4_f32)
#define HAVE_WMMA 1
#else
#define HAVE_WMMA 0
#endif

typedef unsigned int u32x4 __attribute__((ext_vector_type(4)));
typedef int          i32x4 __attribute__((ext_vector_type(4)));
typedef int          i32x8 __attribute__((ext_vector_type(8)));
typedef float        v2f   __attribute__((ext_vector_type(2)));
typedef float        v8f   __attribute__((ext_vector_type(8)));

// Bit-exact "normalize |v|+eps into [0.75, 1.5)" — matches the reference's
// piecewise log2/exp2 map (continuous across exponent rounding).
__device__ __forceinline__ float mant_map(float v) {
  float a = fabsf(v) + EPSF;
  uint32_t b = __float_as_uint(a);
  b = (b & 0x007FFFFFu) | 0x3F800000u;     // mantissa in [1,2)
  float m = __uint_as_float(b);
  return (m >= 1.5f) ? 0.5f * m : m;       // [0.75, 1.5)
}

// CDNA5 async global -> LDS copy (GV mode, per-lane addresses). ASYNCcnt.
__device__ __forceinline__ void async_g2lds_b32(uint32_t lds_byte, const float* g) {
  asm volatile("global_load_async_to_lds_b32 %0, %1, off"
               :
               : "v"(lds_byte), "v"(g)
               : "memory");
}

__device__ __forceinline__ void wait_async0() {
#if __has_builtin(__builtin_amdgcn_s_wait_asynccnt)
  __builtin_amdgcn_s_wait_asynccnt(0);
#else
  asm volatile("s_wait_asynccnt 0" ::: "memory");
#endif
}

#if HAVE_TDM
// TDM: store a dense 56x56 fp32 tile from LDS to global (tile == tensor).
// D# packing per CDNA5 ISA 8.3/8.4.
__device__ __forceinline__ void tdm_store_plane(uint32_t lds_byte, float* gdst) {
  uint64_t ga = ((uint64_t)(uintptr_t)gdst) & ((1ull << 57) - 1ull);
  u32x4 g0;
  g0[0] = 1u;                                   // count=1 (valid, user mode)
  g0[1] = lds_byte;                             // lds_addr (bytes)
  g0[2] = (uint32_t)ga;                         // global_addr[31:0]
  g0[3] = (uint32_t)(ga >> 32) | (2u << 30);    // global_addr[56:32] | type=2
  i32x8 g1;
  g1[0] = (int)(2u << 16);                      // wg_mask=0 | data_size=4B
  g1[1] = (int)((uint32_t)WW << 16);            // tensor_dim0[15:0]=56
  g1[2] = (int)((uint32_t)HH << 16);            // dim0_hi=0 | tensor_dim1[15:0]=56
  g1[3] = (int)((uint32_t)WW << 16);            // dim1_hi=0 | tile_dim0=56
  g1[4] = (int)HH;                              // tile_dim1=56 | tile_dim2=0
  g1[5] = (int)WW;                              // tensor_dim0_stride[31:0]=56
  g1[6] = (int)((uint32_t)INTER << 16);         // stride0_hi=0 | stride1[15:0]=3136
  g1[7] = 0;                                    // stride1_hi=0
  i32x4 gz = {0, 0, 0, 0};
#if defined(__clang_major__) && (__clang_major__ >= 23)
  i32x8 gz8 = {0, 0, 0, 0, 0, 0, 0, 0};
  __builtin_amdgcn_tensor_store_from_lds(g0, g1, gz, gz, gz8, 0);
#else
  __builtin_amdgcn_tensor_store_from_lds(g0, g1, gz, gz, 0);
#endif
}
__device__ __forceinline__ void wait_tensor0() {
#if __has_builtin(__builtin_amdgcn_s_wait_tensorcnt)
  __builtin_amdgcn_s_wait_tensorcnt(0);
#else
  asm volatile("s_wait_tensorcnt 0" ::: "memory");
#endif
}
#endif

__global__ __launch_bounds__(TPB)
void appro_waconv2d_kernel(const float* __restrict__ x,
                           const float* __restrict__ wgt,
                           float* __restrict__ out) {
  // De-interleaved padded planes: xs = x values, ts = x/m1p. Zero halo =>
  // padded taps contribute 0. Separate arrays make each WMMA B fragment a
  // single contiguous ds_load_b64 (no shuffle movs).
  __shared__ __align__(16) float xs[PROWS * PITCH];
  __shared__ __align__(16) float ts[PROWS * PITCH];
  __shared__ float2 uv[KK];               // (.x = w/m2p, .y = (m2p-1)*w/m2p)
#if HAVE_TDM
  __shared__ float ostage[INTER];         // dense output tile for TDM store
#endif

  const int tid   = threadIdx.x;
  const int plane = blockIdx.x;           // b*NC + c
  const int ch    = plane % NC;
  const float* xp = x   + (size_t)plane * INTER;
  float*       op = out + (size_t)plane * INTER;

  // ---- zero LDS (halo + defaults) ----
  for (int i = tid; i < (PROWS * PITCH) / 2; i += TPB) {
    ((float2*)xs)[i] = make_float2(0.0f, 0.0f);
    ((float2*)ts)[i] = make_float2(0.0f, 0.0f);
  }

  // ---- per-channel tap factors:  factor*x*w == x*u + (x/m1p)*v ----
  if (tid < KK) {
    float wv  = wgt[ch * KK + tid];
    float m2p = mant_map(wv);
    float uu  = wv / m2p;
    uv[tid] = make_float2(uu, (m2p - 1.0f) * uu);
  }
  __syncthreads();

  // ---- async-stage the 56x56 x-plane into the padded interior of xs ----
  const uint32_t lds_base = (uint32_t)(uintptr_t)(&xs[0]);
  for (int i = tid; i < INTER; i += TPB) {
    int r = i / WW, cc = i - r * WW;
    uint32_t off = lds_base + (uint32_t)(((r + 3) * PITCH + (cc + 3)) * 4);
    async_g2lds_b32(off, xp + i);
  }
  wait_async0();
  __syncthreads();

  // ---- elementwise: t = x / mant_map(x)  (halo stays 0) ----
  for (int i = tid; i < INTER; i += TPB) {
    int r = i / WW, cc = i - r * WW;
    int idx = (r + 3) * PITCH + (cc + 3);
    float xv = xs[idx];
    ts[idx] = xv / mant_map(xv);
  }
  __syncthreads();

#if HAVE_WMMA
  // =========================================================================
  // WMMA path: out = conv(x,u)+conv(t,v) as Toeplitz-weight matmuls.
  //   C[m][n] += A_i[m][k] * B_i[k][n],  A_i[m][k] = u[i][k-m] (shifted diag),
  //   B_i[k][n] = srcpad[ho_n + i][wob_n + k],  n = 16 (row, col-block) tasks.
  // f32 WMMA 16x16x4, K=24 per i (6 chunks), 7 i-rows, 2 convs = 84 wmma/tile.
  // =========================================================================
  const int lane = tid & 31;
  const int lm   = lane & 15;             // A: m   | B/C: n   (column task)
  const int lh   = lane >> 4;             // K-half (A/B), M-half (C/D)
  const int wid  = tid >> 5;

  // Per-lane A fragments for both convs (A VGPR r holds K = 2*lh + r).
  float Au[7][6][2], Av[7][6][2];
  #pragma unroll
  for (int i = 0; i < 7; ++i) {
    #pragma unroll
    for (int c6 = 0; c6 < 6; ++c6) {
      #pragma unroll
      for (int r = 0; r < 2; ++r) {
        int d = c6 * 4 + lh * 2 + r - lm;        // tap column j = kglob - m
        bool ok = (d >= 0) && (d < 7);
        float2 p = uv[i * 7 + (ok ? d : 0)];
        Au[i][c6][r] = ok ? p.x : 0.0f;
        Av[i][c6][r] = ok ? p.y : 0.0f;
      }
    }
  }

  for (int T = wid; T < NTILE; T += TPB / 32) {
    int q   = T * 16 + lm;                // this lane's column task (N index)
    int ho  = q >> 2;                     // output row
    int blk = q & 3;                      // column block
    int wob = (blk < 3) ? blk * 16 : 40;  // block base (4th overlaps: 40..55)
    int bb  = ho * PITCH + wob + 2 * lh;  // B base (float elements, 8B-aligned)

    v8f C = {0.f, 0.f, 0.f, 0.f, 0.f, 0.f, 0.f, 0.f};
    #pragma unroll
    for (int i = 0; i < 7; ++i) {
      #pragma unroll
      for (int c6 = 0; c6 < 6; ++c6) {
        int bidx = bb + i * PITCH + c6 * 4;          // K = 2*lh .. 2*lh+1
        v2f Bx = *(const v2f*)&xs[bidx];             // one ds_load_b64
        v2f Bt = *(const v2f*)&ts[bidx];             // one ds_load_b64
        v2f Aui = {Au[i][c6][0], Au[i][c6][1]};
        v2f Avi = {Av[i][c6][0], Av[i][c6][1]};
        C = __builtin_amdgcn_wmma_f32_16x16x4_f32(false, Aui, false, Bx,
                                                  (short)0, C, false, false);
        C = __builtin_amdgcn_wmma_f32_16x16x4_f32(false, Avi, false, Bt,
                                                  (short)0, C, false, false);
      }
    }
    // Lane holds C[m][n] for n=lm, m=8*lh+r. Block 3 stores only cols 48..55
    // (its lower half duplicates block 2 -> avoid racy double writes).
    if (blk != 3 || lh == 1) {
      int ob = ho * WW + wob + 8 * lh;
      #pragma unroll
      for (int r = 0; r < 8; ++r) {
#if HAVE_TDM
        ostage[ob + r] = C[r];
#else
        op[ob + r] = C[r];
#endif
      }
    }
  }
#else
  // ======================= VALU fallback path ==============================
  float u[KK], v[KK];
  #pragma unroll
  for (int k = 0; k < KK; ++k) { float2 p = uv[k]; u[k] = p.x; v[k] = p.y; }

  for (int task = tid; task < NTASK; task += TPB) {
    int wo = task % WW;
    int hs = (task / WW) * 4;
    float acc[4] = {0.0f, 0.0f, 0.0f, 0.0f};
    int base = hs * PITCH + wo;
    #pragma unroll
    for (int r2 = 0; r2 < 10; ++r2) {
      float wx[7], wt[7];
      #pragma unroll
      for (int j = 0; j < 7; ++j) {
        wx[j] = xs[base + r2 * PITCH + j];
        wt[j] = ts[base + r2 * PITCH + j];
      }
      #pragma unroll
      for (int a = 0; a < 4; ++a) {
        int i = r2 - a;
        if (i >= 0 && i < 7) {
          #pragma unroll
          for (int j = 0; j < 7; ++j) {
            acc[a] = fmaf(wx[j], u[i * 7 + j], acc[a]);
            acc[a] = fmaf(wt[j], v[i * 7 + j], acc[a]);
          }
        }
      }
    }
    #pragma unroll
    for (int a = 0; a < 4; ++a) {
#if HAVE_TDM
      ostage[(hs + a) * WW + wo] = acc[a];
#else
      op[(hs + a) * WW + wo] = acc[a];
#endif
    }
  }
#endif  // HAVE_WMMA

#if HAVE_TDM
  __syncthreads();                        // ostage fully written (DScnt + barrier)
  if (tid < 32) {                         // single wave issues the DMA
    tdm_store_plane((uint32_t)(uintptr_t)(&ostage[0]), op);
    wait_tensor0();                       // TENSORcnt -> 0 before wave exit
  }
#endif
}

extern "C" void kernel_launch(void* const* d_in, const int* in_sizes, int n_in,
                              void* d_out, int out_size, void* d_ws, size_t ws_size,
                              hipStream_t stream) {
  (void)in_sizes; (void)n_in; (void)out_size; (void)d_ws; (void)ws_size;
  const float* x   = (const float*)d_in[0];   // (4,192,56,56) fp32
  const float* wgt = (const float*)d_in[1];   // (192,1,7,7)   fp32
  float* out = (float*)d_out;                 // (4,192,56,56) fp32
  dim3 grid(4 * NC);   // one workgroup per (batch, channel) plane
  dim3 block(TPB);     // 8 waves (wave32)
  appro_waconv2d_kernel<<<grid, block, 0, stream>>>(x, wgt, out);
}